// DistMult_Layer_26371099197611
// MI455X (gfx1250) — compile-verified
//
#include <hip/hip_runtime.h>
#include <hip/hip_bf16.h>

// DistMult edge scoring on gfx1250 (MI455X).
//   score[e] = sum_d X[src[e],d] * R[type[e],d] * X[dst[e],d]
//
// Strategy: memory/gather-bound (AI ~0.25 FLOP/B, X table is L2-resident).
//  - R (237x64 f32 = 60KB) staged in LDS once per persistent block,
//    rows padded to 66 dwords to avoid 16-way bank conflicts.
//  - 16 edges per wave32 tile; triple product mapped onto chained
//    V_WMMA_F32_16X16X4_F32 (full fp32 precision). Scores = diag(A x B),
//    where A[i,k] = s[i,k]*t[i,k] and B[k,j] = R[type_j, k].
//  - f32 WMMA lane layout (ISA 7.12.2): lane l<16 holds (M=l, K={k,k+1}),
//    lane l+16 holds (M=l, K={k+2,k+3}); B mirrors with N=lane%16. So every
//    lane loads float2 at dim offset k = 4*it + 2*(lane>=16) for s, t and R.
//  - diag(D): D[j,j] sits in lane j / VGPR j (j<8) and lane j+16 / VGPR j-8
//    (j>=8) -> predicated store, no cross-lane ops.
//  - 1,250,000 = 16 * 78,125 exactly: no tail, EXEC stays all-ones (WMMA req).

typedef __attribute__((ext_vector_type(2))) float v2f;
typedef __attribute__((ext_vector_type(8))) float v8f;

#define XDIM       64
#define NUM_REL    237
#define NUM_EDGES  1250000
#define NUM_TILES  (NUM_EDGES / 16)   // 78125
#define R_STRIDE   66                 // 64 + 2 dword pad: bank = (2*type + k) % 64

__global__ __launch_bounds__(256) void distmult_wmma_kernel(
    const float* __restrict__ X,
    const float* __restrict__ R,
    const int*   __restrict__ edge_list,
    const int*   __restrict__ edge_type,
    float*       __restrict__ out)
{
  __shared__ float Rlds[NUM_REL * R_STRIDE];   // 62,568 B

  // Stage relation table into LDS (reads hit L2 after first block).
  for (int i = threadIdx.x; i < NUM_REL * XDIM; i += blockDim.x) {
    const int r = i >> 6, c = i & 63;
    Rlds[r * R_STRIDE + c] = R[i];
  }
  __syncthreads();

  const int lane  = threadIdx.x & 31;
  const int wid   = blockIdx.x * (blockDim.x >> 5) + (threadIdx.x >> 5);
  const int nwv   = gridDim.x * (blockDim.x >> 5);
  const int erow  = lane & 15;            // edge slot within the 16-edge tile
  const int khalf = (lane >> 4) << 1;     // lanes 16..31 cover K offsets +2,+3

  for (int t = wid; t < NUM_TILES; t += nwv) {
    const int e0 = t << 4;
    const int e  = e0 + erow;

    const int src = edge_list[e];
    const int dst = edge_list[NUM_EDGES + e];
    const int typ = edge_type[e];

    // Prefetch next tile's index stream (global_prefetch_b8).
    const int tn = t + nwv;
    if (tn < NUM_TILES) {
      __builtin_prefetch(&edge_list[(tn << 4) + erow], 0, 0);
      __builtin_prefetch(&edge_list[NUM_EDGES + (tn << 4) + erow], 0, 0);
      __builtin_prefetch(&edge_type[(tn << 4) + erow], 0, 0);
    }

    const float* ps = X + (size_t)src * XDIM;
    const float* pt = X + (size_t)dst * XDIM;
    const float* pr = &Rlds[typ * R_STRIDE];

#if __has_builtin(__builtin_amdgcn_wmma_f32_16x16x4_f32)
    v8f acc = {0.f, 0.f, 0.f, 0.f, 0.f, 0.f, 0.f, 0.f};
#pragma unroll
    for (int it = 0; it < 16; ++it) {
      const int k = (it << 2) + khalf;           // 8B-aligned dim offset
      v2f s2 = *(const v2f*)(ps + k);
      v2f t2 = *(const v2f*)(pt + k);
      v2f b2 = *(const v2f*)(pr + k);            // ds_load_b64 from LDS
      v2f a2 = s2 * t2;                          // A[i,k] = s*t
      acc = __builtin_amdgcn_wmma_f32_16x16x4_f32(
          /*neg_a=*/false, a2, /*neg_b=*/false, b2,
          /*c_mod=*/(short)0, acc, /*reuse_a=*/false, /*reuse_b=*/false);
    }
    // Extract diag(D): lane j holds D[j,j] in VGPR j (j<8);
    // lane j+16 holds D[j,j] in VGPR j-8 (j>=8).
    const int j = (lane < 16) ? lane : (lane - 24);
    float d = acc[0];
    if (j == 1) d = acc[1];
    if (j == 2) d = acc[2];
    if (j == 3) d = acc[3];
    if (j == 4) d = acc[4];
    if (j == 5) d = acc[5];
    if (j == 6) d = acc[6];
    if (j == 7) d = acc[7];
    if (lane < 8)        out[e0 + lane]        = d;
    else if (lane >= 24) out[e0 + (lane - 16)] = d;
#else
    // VALU fallback (also satisfies the host-compile pass).
    float partial = 0.f;
#pragma unroll
    for (int it = 0; it < 16; ++it) {
      const int k = (it << 2) + khalf;
      v2f s2 = *(const v2f*)(ps + k);
      v2f t2 = *(const v2f*)(pt + k);
      v2f b2 = *(const v2f*)(pr + k);
      partial += s2[0] * t2[0] * b2[0] + s2[1] * t2[1] * b2[1];
    }
    partial += __shfl_xor(partial, 16, 32);
    if (lane < 16) out[e] = partial;
#endif
  }
}

extern "C" void kernel_launch(void* const* d_in, const int* in_sizes, int n_in,
                              void* d_out, int out_size, void* d_ws, size_t ws_size,
                              hipStream_t stream) {
  (void)in_sizes; (void)n_in; (void)out_size; (void)d_ws; (void)ws_size;
  const float* X         = (const float*)d_in[0];
  const float* R         = (const float*)d_in[1];
  const int*   edge_list = (const int*)d_in[2];
  const int*   edge_type = (const int*)d_in[3];
  float*       out       = (float*)d_out;

  // Persistent grid: 1024 blocks x 256 threads = 8192 wave32 waves,
  // each grid-striding over the 78,125 sixteen-edge tiles. Keeps the
  // 60KB R->LDS preload amortized over ~9.5 tiles per wave.
  distmult_wmma_kernel<<<dim3(1024), dim3(256), 0, stream>>>(
      X, R, edge_list, edge_type, out);
}